// KMeansHRMInnerModule_62852551410250
// MI455X (gfx1250) — compile-verified
//
#include <hip/hip_runtime.h>
#include <hip/hip_bf16.h>

// ---------------------------------------------------------------------------
// KMeans-over-graph, MI455X (gfx1250, wave32).
//
// Algebraic restructuring: segment_sum and the @Wm projection commute, so the
// 3.2M-edge message pass moves 1 scalar/head instead of 128 floats/head:
//   s_k[n]  = relu((x[n]*m[n,k]) @ Ww_k) . Wm_k      (fused fp32 WMMA GEMM)
//   b_k[n]  = (x[n]*m[n,k]) . Wm_k
//   score_k[d] = b_k[d] + sum_{valid e: s->d} s_k[src(e)]
//   head bit = score > 0  (tanh is monotone, tanh(0)=0)
//   output: 1.0 at the first <=2 true head bits (top-2 of equal per-node
//   norms ties-break to ascending head index; norms never needed).
// ---------------------------------------------------------------------------

#define NDIM 128
#define KH   8
#define LDS_STRIDE 132   // pad 128->132 floats: disjoint bank pairs for b64 B-loads

typedef float v2f __attribute__((ext_vector_type(2)));
typedef float v8f __attribute__((ext_vector_type(8)));

// ---------------------------------------------------------------------------
// Kernel 1: per (128-row group, head): fused GEMM + relu + Wm-projection.
// Wave w of the block owns rows [r0, r0+16). Accumulates the full 16x128
// w-tile in 8 f32 WMMA accumulators, never spilling w to memory.
// Ww is staged TRANSPOSED in LDS so each WMMA B-operand (two consecutive
// K-values of one output column) is a single conflict-free ds_load_b64.
// ---------------------------------------------------------------------------
__global__ __launch_bounds__(256) void kmeans_head_gemm(
    const float* __restrict__ x,     // [N,128]
    const float* __restrict__ mask,  // [N,8]
    const float* __restrict__ Ww,    // [8,128,128]
    const float* __restrict__ Wm,    // [8,128]
    float* __restrict__ acc,         // [N,8]  <- base score b (init for edge pass)
    float* __restrict__ sbuf,        // [N,8]  <- per-node message scalar s
    int N)
{
    __shared__ float sWwT[NDIM * LDS_STRIDE];   // 66 KB: sWwT[j*132 + d] = Ww[d][j]
    __shared__ float sWm[NDIM];

    const int head = blockIdx.y;
    const int tid  = threadIdx.x;

    // Stage Ww[head] transposed: coalesced float4 reads along j, scattered
    // LDS writes (once per block; 4096 float4 over 256 threads).
    const float4* gW4 = reinterpret_cast<const float4*>(Ww + (size_t)head * NDIM * NDIM);
#pragma unroll
    for (int it = 0; it < (NDIM * NDIM / 4) / 256; ++it) {
        const int idx = tid + it * 256;        // element in [0, 4096)
        const int d   = idx >> 5;              // source row (K dim)
        const int j0  = (idx & 31) * 4;        // source col (output dim)
        const float4 v = gW4[idx];
        sWwT[(j0 + 0) * LDS_STRIDE + d] = v.x;
        sWwT[(j0 + 1) * LDS_STRIDE + d] = v.y;
        sWwT[(j0 + 2) * LDS_STRIDE + d] = v.z;
        sWwT[(j0 + 3) * LDS_STRIDE + d] = v.w;
    }
    if (tid < NDIM) sWm[tid] = Wm[head * NDIM + tid];
    __syncthreads();

    const int lane = tid & 31;
    const int wv   = tid >> 5;
    const int r0   = blockIdx.x * 128 + wv * 16;   // 16-row tile per wave
    if (r0 >= N) return;                           // wave-uniform: EXEC stays full

    const int mrow = lane & 15;        // row-in-tile for A, col-in-tile for B/C
    const int half = lane >> 4;        // lane half selects K pair {0,1} vs {2,3}
    const int koff = half * 2;
    const int row  = r0 + mrow;
    const float mval = mask[row * KH + head];      // per-row mask scalar (xm = x*m)

    v8f c[8];
#pragma unroll
    for (int ct = 0; ct < 8; ++ct) { v8f z = {}; c[ct] = z; }

    const float* xrow = x + (size_t)row * NDIM;

    // 32 K-steps of V_WMMA_F32_16X16X4_F32 across 8 column tiles.
    // A 16x4 layout: lanes0-15 hold (K=0,1), lanes16-31 hold (K=2,3) of row M=lane%16.
    // B 4x16 layout mirrors the K split across lane halves; each B operand is
    // one aligned b64 LDS load thanks to the transposed staging.
#pragma unroll 2
    for (int ks = 0; ks < 32; ++ks) {
        v2f a = *reinterpret_cast<const v2f*>(xrow + ks * 4 + koff);
        a.x *= mval;
        a.y *= mval;
        const int br = ks * 4 + koff;
#pragma unroll
        for (int ct = 0; ct < 8; ++ct) {
            const v2f b = *reinterpret_cast<const v2f*>(
                &sWwT[(ct * 16 + mrow) * LDS_STRIDE + br]);
            c[ct] = __builtin_amdgcn_wmma_f32_16x16x4_f32(
                /*neg_a=*/false, a, /*neg_b=*/false, b,
                /*c_mod=*/(short)0, c[ct], /*reuse_a=*/false, /*reuse_b=*/false);
        }
    }

    // s[n] = sum_j relu(w[n,j]) * Wm[j].
    // C layout: VGPR v / lanes0-15 -> row M=v, col=lane; lanes16-31 -> row M=v+8.
    float p[8];
#pragma unroll
    for (int v = 0; v < 8; ++v) p[v] = 0.0f;
#pragma unroll
    for (int ct = 0; ct < 8; ++ct) {
        const float wm = sWm[ct * 16 + mrow];
#pragma unroll
        for (int v = 0; v < 8; ++v) {
            const float r = c[ct][v];
            p[v] += (r > 0.0f ? r : 0.0f) * wm;   // select-based relu (no canonicalize)
        }
    }
    // Butterfly within each 16-lane group (xor masks <= 8 stay in-group).
#pragma unroll
    for (int v = 0; v < 8; ++v) {
        p[v] += __shfl_xor(p[v], 1, 32);
        p[v] += __shfl_xor(p[v], 2, 32);
        p[v] += __shfl_xor(p[v], 4, 32);
        p[v] += __shfl_xor(p[v], 8, 32);
    }
    if (lane == 0) {
#pragma unroll
        for (int v = 0; v < 8; ++v) sbuf[(size_t)(r0 + v) * KH + head] = p[v];
    } else if (lane == 16) {
#pragma unroll
        for (int v = 0; v < 8; ++v) sbuf[(size_t)(r0 + 8 + v) * KH + head] = p[v];
    }

    // b[n] = xm[n].Wm  (no relu). Lane pair (L, L+16) splits the 128-dot in half.
    float bsum = 0.0f;
    const int j0 = half * 64;
#pragma unroll 8
    for (int j = 0; j < 64; ++j) bsum += xrow[j0 + j] * sWm[j0 + j];
    bsum *= mval;
    bsum += __shfl_xor(bsum, 16, 32);
    if (half == 0) acc[(size_t)row * KH + head] = bsum;   // base score init
}

// ---------------------------------------------------------------------------
// Kernel 2: pack the 8 per-head membership bits into one word per node.
// ---------------------------------------------------------------------------
__global__ void kmeans_pack_mask(const float* __restrict__ mask,
                                 unsigned int* __restrict__ mb, int N)
{
    int n = blockIdx.x * blockDim.x + threadIdx.x;
    if (n >= N) return;
    unsigned int u = 0;
#pragma unroll
    for (int k = 0; k < KH; ++k)
        u |= (mask[(size_t)n * KH + k] > 0.0f) ? (1u << k) : 0u;
    mb[n] = u;
}

// ---------------------------------------------------------------------------
// Kernel 3: edge pass. acc[dst,k] += s[src,k] for heads where both endpoints
// are masked in. s/acc are L2-resident (3.2 MB each); mb is 400 KB. The edge
// index stream (25.6 MB) is read linearly from HBM -> deep prefetch.
// ---------------------------------------------------------------------------
__global__ void kmeans_edge_pass(const int* __restrict__ ei,       // [2,E]
                                 const unsigned int* __restrict__ mb,
                                 const float* __restrict__ sbuf,
                                 float* __restrict__ acc, int E)
{
    int e = blockIdx.x * blockDim.x + threadIdx.x;
    if (e >= E) return;
    __builtin_prefetch(ei + e + 8192, 0, 0);        // global_prefetch_b8 (src stream)
    __builtin_prefetch(ei + E + e + 8192, 0, 0);    // global_prefetch_b8 (dst stream)
    const int src = ei[e];
    const int dst = ei[E + e];
    unsigned int v = mb[src] & mb[dst];
    while (v) {
        const int k = __ffs(v) - 1;
        v &= v - 1;
        atomicAdd(&acc[(size_t)dst * KH + k], sbuf[(size_t)src * KH + k]);
    }
}

// ---------------------------------------------------------------------------
// Kernel 4: finalize. tanh(score)>0 <=> score>0. Per node all true heads share
// the same weight (norm(x)), so top-2 == first <=2 true heads in index order.
// ---------------------------------------------------------------------------
__global__ void kmeans_finalize(const float* __restrict__ acc,
                                float* __restrict__ out, int N)
{
    int n = blockIdx.x * blockDim.x + threadIdx.x;
    if (n >= N) return;
    int cnt = 0;
#pragma unroll
    for (int k = 0; k < KH; ++k) {
        const bool t = acc[(size_t)n * KH + k] > 0.0f;
        out[(size_t)n * KH + k] = (t && cnt < 2) ? 1.0f : 0.0f;
        cnt += t ? 1 : 0;
    }
}

// ---------------------------------------------------------------------------
extern "C" void kernel_launch(void* const* d_in, const int* in_sizes, int n_in,
                              void* d_out, int out_size, void* d_ws, size_t ws_size,
                              hipStream_t stream)
{
    const float* x    = (const float*)d_in[0];   // [N,128]
    const int*   ei   = (const int*)d_in[1];     // [2,E]
    const float* mask = (const float*)d_in[2];   // [N,8]
    const float* Ww   = (const float*)d_in[3];   // [8,128,128]
    const float* Wm   = (const float*)d_in[4];   // [8,128,1]

    const int N = in_sizes[0] / NDIM;
    const int E = in_sizes[1] / 2;

    // Workspace: acc[N*8] | s[N*8] | mb[N]  (~6.8 MB)
    float* acc  = (float*)d_ws;
    float* sbuf = acc + (size_t)N * KH;
    unsigned int* mb = (unsigned int*)(sbuf + (size_t)N * KH);

    dim3 g1((N + 127) / 128, KH);
    kmeans_head_gemm<<<g1, 256, 0, stream>>>(x, mask, Ww, Wm, acc, sbuf, N);
    kmeans_pack_mask<<<(N + 255) / 256, 256, 0, stream>>>(mask, mb, N);
    kmeans_edge_pass<<<(E + 255) / 256, 256, 0, stream>>>(ei, mb, sbuf, acc, E);
    kmeans_finalize<<<(N + 255) / 256, 256, 0, stream>>>(acc, (float*)d_out, N);
}